// Removal_2345052143700
// MI455X (gfx1250) — compile-verified
//
#include <hip/hip_runtime.h>

// Reference collapse proof:
//   softmax over the singleton channel axis => probs == 1.0 exactly (finite scores).
//   lax.top_k is stable on ties => top_idx = arange(384) per row, top_vals = 1.0.
// Outputs: ids_out = input_ids[:, :384], mask_out = attention_mask[:, :384],
//          probs_sum = 384.0 per row.
// => Optimal MI455X kernel is a 0-FLOP, ~0.4 MB copy (launch-latency bound).
//    Any WMMA/conv work would be dead compute discarded by the singleton softmax.

namespace {
constexpr int B_ROWS   = 64;
constexpr int L_SEQ    = 1024;
constexpr int K_TOP    = 384;
constexpr int V4_ROW_I = L_SEQ / 4;  // 256 int4 per input row
constexpr int V4_ROW_O = K_TOP / 4;  // 96  int4 per output row
}

__global__ __launch_bounds__(128) void removal_collapse_copy(
    const int4* __restrict__ ids_in,    // (64, 1024) int32 viewed as int4
    const int4* __restrict__ mask_in,   // (64, 1024) int32 viewed as int4
    int4*       __restrict__ ids_out,   // (64, 384)  int32 viewed as int4
    int4*       __restrict__ mask_out,  // (64, 384)  int32 viewed as int4
    float*      __restrict__ probs_sum) // (64,)
{
    const int row = blockIdx.x;          // one block per batch row
    const int j   = threadIdx.x;         // 0..127 (4 wave32s)

    if (j < V4_ROW_O) {
        // 16-byte vector copies: global_load_b128 -> global_store_b128
        const int src = row * V4_ROW_I + j;
        const int dst = row * V4_ROW_O + j;
        ids_out[dst]  = ids_in[src];
        mask_out[dst] = mask_in[src];
    } else if (j == V4_ROW_O) {
        // sum of 384 top_vals, each exactly 1.0
        probs_sum[row] = 384.0f;
    }
}

extern "C" void kernel_launch(void* const* d_in, const int* in_sizes, int n_in,
                              void* d_out, int out_size, void* d_ws, size_t ws_size,
                              hipStream_t stream) {
    (void)in_sizes; (void)n_in; (void)out_size; (void)d_ws; (void)ws_size;

    // setup_inputs() order: input_ids, attention_mask, emb_table, conv_w, conv_b.
    // emb_table / conv_w / conv_b are provably unused (singleton softmax).
    const int4* ids_in  = reinterpret_cast<const int4*>(d_in[0]);
    const int4* mask_in = reinterpret_cast<const int4*>(d_in[1]);

    // d_out flat layout (all 4-byte elements):
    //   [0, 24576)        ids_out   int32
    //   [24576, 49152)    mask_out  int32
    //   [49152, 49216)    probs_sum float32
    int* out_base = reinterpret_cast<int*>(d_out);
    int4*  ids_out   = reinterpret_cast<int4*>(out_base);
    int4*  mask_out  = reinterpret_cast<int4*>(out_base + B_ROWS * K_TOP);
    float* probs_sum = reinterpret_cast<float*>(out_base + 2 * B_ROWS * K_TOP);

    removal_collapse_copy<<<B_ROWS, 128, 0, stream>>>(
        ids_in, mask_in, ids_out, mask_out, probs_sum);
}